// GivensRotations_48326972014896
// MI455X (gfx1250) — compile-verified
//
#include <hip/hip_runtime.h>
#include <hip/hip_bf16.h>

typedef __attribute__((ext_vector_type(2))) float v2f;
typedef __attribute__((ext_vector_type(8))) float v8f;

constexpr int HD       = 64;                  // head dim == GEMM N == GEMM K
constexpr int NROT     = 128;                 // number of Givens rotations
constexpr long ROWS    = 8L * 4096 * 16;      // B*S*nH = 524288 rows
constexpr int WAVES_PB = 8;                   // 256 threads / wave32
constexpr int BLOCKS   = 1024;
constexpr long SLABS   = ROWS / 16;           // 32768 16-row slabs
constexpr int SPW      = (int)(SLABS / ((long)BLOCKS * WAVES_PB)); // 4 slabs/wave
constexpr int PAD      = 68;                  // row stride in dwords: 16B aligned, conflict-free

// ---------------------------------------------------------------------------
// Kernel 1: fold 128 Givens rotations into a single 64x64 matrix, store M^T.
// Thread t owns row t of M (held in LDS, padded to avoid 64-way bank conflicts).
// M_{k+1} = M_k * G_k  =>  col_i' = c*col_i + s*col_j ; col_j' = -s*col_i + c*col_j
// ---------------------------------------------------------------------------
__global__ __launch_bounds__(64) void build_m_kernel(const float* __restrict__ thetas,
                                                     float* __restrict__ mt) {
  __shared__ float M[HD][HD + 1];
  const int t = threadIdx.x;
  for (int c = 0; c < HD; ++c) M[t][c] = (t == c) ? 1.0f : 0.0f;
  for (int k = 0; k < NROT; ++k) {
    const int i = k & (HD - 1);
    const int j = (k + 1) & (HD - 1);
    float s, c;
    sincosf(thetas[k], &s, &c);
    const float mi = M[t][i];
    const float mj = M[t][j];
    M[t][i] = c * mi + s * mj;
    M[t][j] = -s * mi + c * mj;
  }
  // mt[n*64 + k] = M[k][n]  (transposed: contiguous in K for the WMMA B feed)
  for (int n = 0; n < HD; ++n) mt[n * HD + t] = M[t][n];
}

// ---------------------------------------------------------------------------
// Kernel 2: out = x @ M via V_WMMA_F32_16X16X4_F32 (exact fp32 math).
// One wave per 16-row slab; 4 N-tiles of 16x16 accumulated over K=64 (16 steps).
// ---------------------------------------------------------------------------
__global__ __launch_bounds__(256) void givens_gemm_kernel(const float* __restrict__ x,
                                                          const float* __restrict__ mt,
                                                          float* __restrict__ out) {
  __shared__ float Bt[HD][PAD];            // Bt[n][k] = M[k][n]
  __shared__ float As[WAVES_PB][16][PAD];  // per-wave x staging (16 rows x 64 k)

  for (int t = threadIdx.x; t < HD * HD; t += 256)
    Bt[t >> 6][t & 63] = mt[t];
  __syncthreads();

  const int wave  = threadIdx.x >> 5;
  const int lane  = threadIdx.x & 31;
  const int l16   = lane & 15;
  const int lhalf = lane >> 4;  // 0: lanes 0-15, 1: lanes 16-31

  const long waveGlobal = (long)blockIdx.x * WAVES_PB + wave;

  for (int it = 0; it < SPW; ++it) {
    const long base = (waveGlobal * SPW + it) * 16;     // first row of this slab
    const float4* xs = (const float4*)(x + base * HD);  // 4KB contiguous slab

    // Stage slab: 8 coalesced b128 loads -> padded LDS rows.
    // dword offset t*128 + lane*4 => row = 2t + lhalf, col = l16*4 (16B aligned).
#pragma unroll
    for (int t = 0; t < 8; ++t) {
      float4 v = xs[t * 32 + lane];
      *(float4*)&As[wave][t * 2 + lhalf][l16 * 4] = v;
    }
    if (it + 1 < SPW) {  // prefetch next slab (32 lanes x 128B = 4KB)
      __builtin_prefetch((const char*)(x + (base + 16) * HD) + lane * 128, 0, 1);
    }
    __syncthreads();

    v8f acc0 = {}, acc1 = {}, acc2 = {}, acc3 = {};
#pragma unroll
    for (int k0 = 0; k0 < HD; k0 += 4) {
      const int kk = k0 + 2 * lhalf;  // A/B ISA layout: half-wave splits the K window
      v2f a  = *(const v2f*)&As[wave][l16][kk];
      v2f b0 = *(const v2f*)&Bt[ 0 + l16][kk];
      v2f b1 = *(const v2f*)&Bt[16 + l16][kk];
      v2f b2 = *(const v2f*)&Bt[32 + l16][kk];
      v2f b3 = *(const v2f*)&Bt[48 + l16][kk];
      acc0 = __builtin_amdgcn_wmma_f32_16x16x4_f32(false, a, false, b0, (short)0, acc0, false, false);
      acc1 = __builtin_amdgcn_wmma_f32_16x16x4_f32(false, a, false, b1, (short)0, acc1, false, false);
      acc2 = __builtin_amdgcn_wmma_f32_16x16x4_f32(false, a, false, b2, (short)0, acc2, false, false);
      acc3 = __builtin_amdgcn_wmma_f32_16x16x4_f32(false, a, false, b3, (short)0, acc3, false, false);
    }

    // C/D layout: VGPR r, lanes 0-15 -> row r, lanes 16-31 -> row r+8; N = l16.
    float* orow = out + (base + 8 * lhalf) * HD + l16;
#pragma unroll
    for (int r = 0; r < 8; ++r) {
      float* p = orow + (long)r * HD;
      __builtin_nontemporal_store(acc0[r], p);
      __builtin_nontemporal_store(acc1[r], p + 16);
      __builtin_nontemporal_store(acc2[r], p + 32);
      __builtin_nontemporal_store(acc3[r], p + 48);
    }
  }
}

extern "C" void kernel_launch(void* const* d_in, const int* in_sizes, int n_in,
                              void* d_out, int out_size, void* d_ws, size_t ws_size,
                              hipStream_t stream) {
  const float* x      = (const float*)d_in[0];
  const float* thetas = (const float*)d_in[1];
  float* out = (float*)d_out;
  float* mt  = (float*)d_ws;  // 64*64*4 = 16KB of workspace for M^T

  build_m_kernel<<<1, 64, 0, stream>>>(thetas, mt);
  givens_gemm_kernel<<<BLOCKS, 256, 0, stream>>>(x, mt, out);
}